// WeightedLoss_1580547967157
// MI455X (gfx1250) — compile-verified
//
#include <hip/hip_runtime.h>
#include <stdint.h>
#include <stddef.h>

// Problem geometry (fixed by the reference: mask[64, 480, 640], 32 classes)
#define NUM_CLASSES 32
#define IMGS        64
#define PIX         307200                 // 480*640
#define F4_PER_IMG  (PIX / 4)              // 76800 float4 per image
#define BPI         75                     // blocks per image
#define F4_PER_BLK  (F4_PER_IMG / BPI)     // 1024 float4 per block
#define THREADS     256                    // 8 waves (wave32)
#define ITERS       (F4_PER_BLK / THREADS) // 4
#define SUBH        32                     // replicated sub-histograms
#define SUBSTRIDE   33                     // pad to kill LDS bank conflicts

// ---- CDNA5 async global->LDS path (guarded; falls back to plain loads) ----
typedef __attribute__((ext_vector_type(4))) int v4i;
typedef __attribute__((address_space(1))) v4i gv4i;   // global (AS1) 16B vector
typedef __attribute__((address_space(3))) v4i lv4i;   // LDS   (AS3) 16B vector

#if defined(__has_builtin)
#  if __has_builtin(__builtin_amdgcn_global_load_async_to_lds_b128)
#    define HAVE_ASYNC_B128 1
#  endif
#endif
#ifndef HAVE_ASYNC_B128
#  define HAVE_ASYNC_B128 0
#endif

#if defined(__has_builtin) && __has_builtin(__builtin_amdgcn_s_wait_asynccnt)
#  define WAIT_ASYNC(n) do { __builtin_amdgcn_s_wait_asynccnt(n); \
                             asm volatile("" ::: "memory"); } while (0)
#else
#  define WAIT_ASYNC(n) asm volatile("s_wait_asynccnt %0" :: "n"(n) : "memory")
#endif

#if HAVE_ASYNC_B128
__device__ __forceinline__ void async_copy16(const float4* g, float4* l) {
  // GLOBAL_LOAD_ASYNC_TO_LDS_B128: per-lane 16B memory->LDS, tracked by ASYNCcnt
  __builtin_amdgcn_global_load_async_to_lds_b128((gv4i*)g, (lv4i*)l,
                                                 /*offset=*/0, /*cpol=*/0);
}
#endif

// ---------------- Pass 0: zero per-image count table in workspace ----------
__global__ void k_zero(unsigned* __restrict__ p, int n) {
  int i = blockIdx.x * blockDim.x + threadIdx.x;
  if (i < n) p[i] = 0u;
}

// ---------------- Pass 1: per-image histogram --------------------------------
// grid = (BPI, IMGS), block = THREADS. Replicated LDS sub-histograms, async-
// double-buffered tile staging, then one global atomic per class per block.
__global__ __launch_bounds__(THREADS) void k_hist(const float* __restrict__ mask,
                                                  unsigned* __restrict__ counts) {
  __shared__ unsigned sub[SUBH * SUBSTRIDE];
#if HAVE_ASYNC_B128
  __shared__ float4 stage[2][THREADS];
#endif
  const int t   = threadIdx.x;
  const int img = blockIdx.y;
  const float4* src = (const float4*)(mask + (size_t)img * PIX)
                      + (size_t)blockIdx.x * F4_PER_BLK;

  for (int i = t; i < SUBH * SUBSTRIDE; i += THREADS) sub[i] = 0u;
  __syncthreads();

  // lane t uses sub-histogram (t & 31); stride 33 => per-lane distinct banks
  unsigned* myh = &sub[(t & (SUBH - 1)) * SUBSTRIDE];

#if HAVE_ASYNC_B128
  async_copy16(src + t, &stage[0][t]);           // prologue: stage tile 0
  for (int it = 0; it < ITERS; ++it) {
    if (it + 1 < ITERS) {
      async_copy16(src + (it + 1) * THREADS + t, &stage[(it + 1) & 1][t]);
      WAIT_ASYNC(1);                             // current tile done, next in flight
    } else {
      WAIT_ASYNC(0);
    }
    float4 v = stage[it & 1][t];                 // each lane reads only its own bytes
    atomicAdd(&myh[((int)v.x) & 31], 1u);
    atomicAdd(&myh[((int)v.y) & 31], 1u);
    atomicAdd(&myh[((int)v.z) & 31], 1u);
    atomicAdd(&myh[((int)v.w) & 31], 1u);
  }
#else
  for (int it = 0; it < ITERS; ++it) {
    float4 v = src[it * THREADS + t];
    atomicAdd(&myh[((int)v.x) & 31], 1u);
    atomicAdd(&myh[((int)v.y) & 31], 1u);
    atomicAdd(&myh[((int)v.z) & 31], 1u);
    atomicAdd(&myh[((int)v.w) & 31], 1u);
  }
#endif
  __syncthreads();

  if (t < NUM_CLASSES) {
    unsigned s = 0;
    #pragma unroll
    for (int h = 0; h < SUBH; ++h) s += sub[h * SUBSTRIDE + t];
    if (s) atomicAdd(&counts[img * NUM_CLASSES + t], s);
  }
}

// ---------------- Pass 2: gather 1/count back to pixels ----------------------
// grid = (BPI, IMGS). Mask (78.6MB) + counts fit in 192MB L2 => reads hit L2.
__global__ __launch_bounds__(THREADS) void k_gather(const float* __restrict__ mask,
                                                    const unsigned* __restrict__ counts,
                                                    float* __restrict__ out) {
  __shared__ float inv[NUM_CLASSES];
  const int t   = threadIdx.x;
  const int img = blockIdx.y;
  if (t < NUM_CLASSES) {
    unsigned c = counts[img * NUM_CLASSES + t];
    inv[t] = c ? (1.0f / (float)c) : 0.0f;
  }
  __syncthreads();

  const float4* src = (const float4*)(mask + (size_t)img * PIX)
                      + (size_t)blockIdx.x * F4_PER_BLK;
  float4* dst = (float4*)(out + (size_t)img * PIX)
                + (size_t)blockIdx.x * F4_PER_BLK;

  #pragma unroll
  for (int it = 0; it < ITERS; ++it) {
    if (it + 1 < ITERS)
      __builtin_prefetch(src + (it + 1) * THREADS + t, 0, 0); // global_prefetch_b8
    float4 v = src[it * THREADS + t];
    float4 w;
    w.x = inv[((int)v.x) & 31];
    w.y = inv[((int)v.y) & 31];
    w.z = inv[((int)v.z) & 31];
    w.w = inv[((int)v.w) & 31];
    dst[it * THREADS + t] = w;
  }
}

extern "C" void kernel_launch(void* const* d_in, const int* in_sizes, int n_in,
                              void* d_out, int out_size, void* d_ws, size_t ws_size,
                              hipStream_t stream) {
  (void)in_sizes; (void)n_in; (void)out_size; (void)ws_size;
  const float* mask   = (const float*)d_in[0];
  float*       out    = (float*)d_out;
  unsigned*    counts = (unsigned*)d_ws;       // IMGS*NUM_CLASSES = 2048 u32 (8KB)

  const int ncounts = IMGS * NUM_CLASSES;
  k_zero<<<(ncounts + THREADS - 1) / THREADS, THREADS, 0, stream>>>(counts, ncounts);
  k_hist<<<dim3(BPI, IMGS), THREADS, 0, stream>>>(mask, counts);
  k_gather<<<dim3(BPI, IMGS), THREADS, 0, stream>>>(mask, counts, out);
}